// AcidSynthLearnedBiquadCoeff_47253230190750
// MI455X (gfx1250) — compile-verified
//
#include <hip/hip_runtime.h>
#include <hip/hip_bf16.h>
#include <cstdint>
#include <cstddef>

// AcidSynth learned-biquad: time-varying order-2 IIR + FIR3 with coefficient
// interpolation. Bandwidth-bound (~534 MB @ 23.3 TB/s ~= 23 us floor).
// - IIR parallelized as an affine (2x2) chunked scan (256 chunks x 128 samples).
// - Logits staged to LDS via CDNA5 async global->LDS + s_wait_asynccnt.
// - Hot loops use register-resident coefficient frames (chunk spans <= 2 frames).
// - All outputs written with non-temporal stores so x stays L2-resident for pass 2.

#define T_DIM   32768
#define F_DIM   128
#define NLOG    5
#define THREADS 256
#define CHUNK   (T_DIM / THREADS)   // 128 samples per thread

typedef float v4f __attribute__((ext_vector_type(4)));   // native vector: NT-store OK

__device__ __forceinline__ unsigned lds_off(const void* p) {
  // Flat shared-aperture addresses carry the LDS byte offset in the low 32 bits.
  return (unsigned)(uintptr_t)p;
}

__global__ __launch_bounds__(THREADS)
void acid_biquad_scan_kernel(const float* __restrict__ x,
                             const float* __restrict__ logits,
                             float* __restrict__ out,
                             int n_rows) {
  const int b   = blockIdx.x;
  const int tid = threadIdx.x;

  __shared__ __align__(16) float s_raw[F_DIM * NLOG];          // raw logits (640 f)
  __shared__ float sA1[F_DIM], sA2[F_DIM];
  __shared__ float sB0[F_DIM], sB1[F_DIM], sB2[F_DIM];
  __shared__ float sU0[THREADS], sU1[THREADS];                 // chunk offsets u
  __shared__ float sP00[THREADS], sP01[THREADS];               // chunk transition P
  __shared__ float sP10[THREADS], sP11[THREADS];

  // ---------------- Phase A: async-stage logits row into LDS (CDNA5 async path)
  {
    const float* grow = logits + (size_t)b * (F_DIM * NLOG);
    if (tid < (F_DIM * NLOG) / 4) {                            // 160 lanes x 16 B
      unsigned           ldst = lds_off(&s_raw[tid * 4]);
      unsigned long long gsrc = (unsigned long long)(uintptr_t)(grow + tid * 4);
      asm volatile("global_load_async_to_lds_b128 %0, %1, off"
                   :: "v"(ldst), "v"(gsrc) : "memory");
    }
#if __has_builtin(__builtin_amdgcn_s_wait_asynccnt)
    __builtin_amdgcn_s_wait_asynccnt(0);
#else
    asm volatile("s_wait_asynccnt 0x0" ::: "memory");
#endif
  }
  __syncthreads();

  // ---------------- Phase B: per-frame stability-triangle transform (tanh) once
  if (tid < F_DIM) {
    const float sf = 1.0f - 0.001f;                            // 1 - STABILITY_EPS
    float l0 = s_raw[tid * NLOG + 0];
    float l1 = s_raw[tid * NLOG + 1];
    float a1 = 2.0f * tanhf(l0) * sf;
    float aa = fabsf(a1);
    float a2 = 0.5f * ((2.0f - aa) * tanhf(l1) * sf + aa);
    sA1[tid] = a1;
    sA2[tid] = a2;
    sB0[tid] = s_raw[tid * NLOG + 2];
    sB1[tid] = s_raw[tid * NLOG + 3];
    sB2[tid] = s_raw[tid * NLOG + 4];
  }
  __syncthreads();

  const float R = 127.0f / 32767.0f;                           // align_corners step
  const float* xrow = x + (size_t)b * T_DIM;
  const int    c0   = tid * CHUNK;

  // Warm this thread's x chunk (512 B) into cache while we stream writes.
  #pragma unroll
  for (int pf = 0; pf < CHUNK * 4; pf += 128)
    __builtin_prefetch((const char*)(xrow + c0) + pf, 0, 0);

  // Register-resident coefficient frames for this thread's chunk.
  // pos range across the chunk is < 0.5 frame wide, so lo in {fs, fs+1}.
  const int fs = [&] {
    int f = (int)((float)c0 * R);
    return f > F_DIM - 2 ? F_DIM - 2 : f;
  }();
  const int f1 = fs + 1;                                       // <= 127
  const int f2 = (fs + 2 > F_DIM - 1) ? (F_DIM - 1) : (fs + 2);
  const float a1r0 = sA1[fs], a1r1 = sA1[f1], a1r2 = sA1[f2];
  const float a2r0 = sA2[fs], a2r1 = sA2[f1], a2r2 = sA2[f2];
  const float b0r0 = sB0[fs], b0r1 = sB0[f1], b0r2 = sB0[f2];
  const float b1r0 = sB1[fs], b1r1 = sB1[f1], b1r2 = sB1[f2];
  const float b2r0 = sB2[fs], b2r1 = sB2[f1], b2r2 = sB2[f2];

  // ---------------- Phase C: coalesced streaming (NT) writes of a_full / b_coeff
  {
    float* aF = out + (size_t)n_rows * T_DIM;                  // [B,T,3]
    float* bC = out + (size_t)4 * n_rows * T_DIM;              // [B,T,3]
    for (int t = tid; t < T_DIM; t += THREADS) {
      float pos = (float)t * R;
      int   f   = (int)pos; f = (f > F_DIM - 2) ? (F_DIM - 2) : f;
      float w   = pos - (float)f;
      float a1  = fmaf(w, sA1[f + 1] - sA1[f], sA1[f]);
      float a2  = fmaf(w, sA2[f + 1] - sA2[f], sA2[f]);
      float b0  = fmaf(w, sB0[f + 1] - sB0[f], sB0[f]);
      float b1  = fmaf(w, sB1[f + 1] - sB1[f], sB1[f]);
      float b2  = fmaf(w, sB2[f + 1] - sB2[f], sB2[f]);
      size_t o = ((size_t)b * T_DIM + t) * 3;
      __builtin_nontemporal_store(1.0f, aF + o + 0);
      __builtin_nontemporal_store(a1,   aF + o + 1);
      __builtin_nontemporal_store(a2,   aF + o + 2);
      __builtin_nontemporal_store(b0,   bC + o + 0);
      __builtin_nontemporal_store(b1,   bC + o + 1);
      __builtin_nontemporal_store(b2,   bC + o + 2);
    }
  }

  // Per-sample coefficient fetch from registers (branchless frame select).
  auto coeffA = [&](int n, float& a1, float& a2) {
    float pos = (float)n * R;
    int   li  = (int)pos; li = (li > F_DIM - 2) ? (F_DIM - 2) : li;
    float w   = pos - (float)li;
    bool  hi  = (li != fs);
    float a1l = hi ? a1r1 : a1r0, a1h = hi ? a1r2 : a1r1;
    float a2l = hi ? a2r1 : a2r0, a2h = hi ? a2r2 : a2r1;
    a1 = fmaf(w, a1h - a1l, a1l);
    a2 = fmaf(w, a2h - a2l, a2l);
  };

  // ---------------- Phase D: pass 1 — chunk-local solution + 2x2 transition
  // state s[n] = [y[n], y[n-1]];  s = M s_prev + e0*x,  M = [[-a1,-a2],[1,0]]
  float u0 = 0.0f, u1 = 0.0f;                                  // local y, zero init
  float p00 = 1.0f, p01 = 0.0f, p10 = 0.0f, p11 = 0.0f;        // P = product of M
  {
    const float4* xq = reinterpret_cast<const float4*>(xrow + c0);
    for (int q = 0; q < CHUNK / 4; ++q) {
      float4 xv = xq[q];
      float xs[4] = {xv.x, xv.y, xv.z, xv.w};
      #pragma unroll
      for (int j = 0; j < 4; ++j) {
        float a1, a2;
        coeffA(c0 + q * 4 + j, a1, a2);
        float y = xs[j] - a1 * u0 - a2 * u1;
        u1 = u0; u0 = y;
        float t0 = -a1 * p00 - a2 * p10;
        float t1 = -a1 * p01 - a2 * p11;
        p10 = p00; p11 = p01; p00 = t0; p01 = t1;
      }
    }
  }
  sU0[tid] = u0;  sU1[tid] = u1;
  sP00[tid] = p00; sP01[tid] = p01; sP10[tid] = p10; sP11[tid] = p11;
  __syncthreads();

  // ---------------- Phase E: Hillis-Steele inclusive scan of affine transforms
  for (int off = 1; off < THREADS; off <<= 1) {
    float au0 = 0, au1 = 0, ap00 = 0, ap01 = 0, ap10 = 0, ap11 = 0;
    float cu0 = 0, cu1 = 0, cp00 = 0, cp01 = 0, cp10 = 0, cp11 = 0;
    const bool act = (tid >= off);
    if (act) {
      au0 = sU0[tid - off];  au1 = sU1[tid - off];
      ap00 = sP00[tid - off]; ap01 = sP01[tid - off];
      ap10 = sP10[tid - off]; ap11 = sP11[tid - off];
      cu0 = sU0[tid];  cu1 = sU1[tid];
      cp00 = sP00[tid]; cp01 = sP01[tid];
      cp10 = sP10[tid]; cp11 = sP11[tid];
    }
    __syncthreads();
    if (act) {
      // later (c) after earlier (a): u' = c_u + c_P a_u ; P' = c_P a_P
      sU0[tid]  = cu0 + cp00 * au0 + cp01 * au1;
      sU1[tid]  = cu1 + cp10 * au0 + cp11 * au1;
      sP00[tid] = cp00 * ap00 + cp01 * ap10;
      sP01[tid] = cp00 * ap01 + cp01 * ap11;
      sP10[tid] = cp10 * ap00 + cp11 * ap10;
      sP11[tid] = cp10 * ap01 + cp11 * ap11;
    }
    __syncthreads();
  }

  // Global initial state is zero, so exact chunk-entry state = u of previous slot.
  float y1 = 0.0f, y2 = 0.0f;
  if (tid > 0) { y1 = sU0[tid - 1]; y2 = sU1[tid - 1]; }

  // ---------------- Phase F: pass 2 — exact recurrence + fused FIR3, NT-write y_ab
  {
    const float4* xq = reinterpret_cast<const float4*>(xrow + c0);
    v4f* yq = reinterpret_cast<v4f*>(out + (size_t)b * T_DIM + c0);
    for (int q = 0; q < CHUNK / 4; ++q) {
      float4 xv = xq[q];
      float xs[4] = {xv.x, xv.y, xv.z, xv.w};
      float rs[4];
      #pragma unroll
      for (int j = 0; j < 4; ++j) {
        int   n   = c0 + q * 4 + j;
        float pos = (float)n * R;
        int   li  = (int)pos; li = (li > F_DIM - 2) ? (F_DIM - 2) : li;
        float w   = pos - (float)li;
        bool  hi  = (li != fs);
        float a1l = hi ? a1r1 : a1r0, a1h = hi ? a1r2 : a1r1;
        float a2l = hi ? a2r1 : a2r0, a2h = hi ? a2r2 : a2r1;
        float b0l = hi ? b0r1 : b0r0, b0h = hi ? b0r2 : b0r1;
        float b1l = hi ? b1r1 : b1r0, b1h = hi ? b1r2 : b1r1;
        float b2l = hi ? b2r1 : b2r0, b2h = hi ? b2r2 : b2r1;
        float a1 = fmaf(w, a1h - a1l, a1l);
        float a2 = fmaf(w, a2h - a2l, a2l);
        float b0 = fmaf(w, b0h - b0l, b0l);
        float b1 = fmaf(w, b1h - b1l, b1l);
        float b2 = fmaf(w, b2h - b2l, b2l);
        float y = xs[j] - a1 * y1 - a2 * y2;
        rs[j] = b0 * y + b1 * y1 + b2 * y2;                    // FIR sees y[n-1],y[n-2]
        y2 = y1; y1 = y;
      }
      v4f r = {rs[0], rs[1], rs[2], rs[3]};
      __builtin_nontemporal_store(r, yq + q);
    }
  }
}

extern "C" void kernel_launch(void* const* d_in, const int* in_sizes, int n_in,
                              void* d_out, int out_size, void* d_ws, size_t ws_size,
                              hipStream_t stream) {
  (void)n_in; (void)out_size; (void)d_ws; (void)ws_size;
  const float* x      = (const float*)d_in[0];   // [B, 32768]
  const float* logits = (const float*)d_in[1];   // [B, 128, 5]
  float*       out    = (float*)d_out;           // y_ab | a_full | b_coeff (flat)
  const int n_rows = in_sizes[0] / T_DIM;        // 512
  acid_biquad_scan_kernel<<<dim3(n_rows), dim3(THREADS), 0, stream>>>(x, logits, out, n_rows);
}